// EdgeConvModule_64312840290829
// MI455X (gfx1250) — compile-verified
//
#include <hip/hip_runtime.h>

typedef __attribute__((ext_vector_type(2))) float v2f;
typedef __attribute__((ext_vector_type(4))) float v4f;
typedef __attribute__((ext_vector_type(8))) float v8f;
typedef __attribute__((ext_vector_type(4))) int   v4i;

#define N_B    32
#define P_PTS  1024
#define K_NN   16
#define C0     64
#define BN_EPS 1e-3f
#define ALPHA  0.1f

#define AS1 __attribute__((address_space(1)))
#define AS3 __attribute__((address_space(3)))

#if __has_builtin(__builtin_amdgcn_global_load_async_to_lds_b128) && \
    __has_builtin(__builtin_amdgcn_s_wait_asynccnt)
#define USE_ASYNC_LDS 1
#else
#define USE_ASYNC_LDS 0
#endif

// ---------------------------------------------------------------------------
// 1) KNN: one block per (n,p). Distances in LDS, 17 stable argmin rounds.
// ---------------------------------------------------------------------------
__global__ void knn_kernel(const float* __restrict__ pts, int* __restrict__ idxOut) {
    const int np = blockIdx.x;            // n*P + p
    const int n  = np >> 10;
    __shared__ float dist[P_PTS];
    __shared__ unsigned long long red[128];

    const float qx = pts[np * 3 + 0];
    const float qy = pts[np * 3 + 1];
    const float qz = pts[np * 3 + 2];

    for (int j = threadIdx.x; j < P_PTS; j += 128) {
        const float* pj = pts + ((size_t)n * P_PTS + j) * 3;
        float dx = qx - pj[0], dy = qy - pj[1], dz = qz - pj[2];
        dist[j] = dx * dx + dy * dy + dz * dz;
    }
    __syncthreads();

    for (int r = 0; r < K_NN + 1; ++r) {
        unsigned long long best = ~0ull;
        for (int j = threadIdx.x; j < P_PTS; j += 128) {
            unsigned long long key =
                ((unsigned long long)__float_as_uint(dist[j]) << 32) | (unsigned)j;
            best = (key < best) ? key : best;
        }
        red[threadIdx.x] = best;
        __syncthreads();
        for (int s = 64; s > 0; s >>= 1) {
            if (threadIdx.x < s) {
                unsigned long long o = red[threadIdx.x + s];
                if (o < red[threadIdx.x]) red[threadIdx.x] = o;
            }
            __syncthreads();
        }
        if (threadIdx.x == 0) {
            int j = (int)(red[0] & 0xFFFFFFFFu);
            if (r > 0) idxOut[np * K_NN + (r - 1)] = j;
            dist[j] = __uint_as_float(0x7F800000u);   // +inf: remove from pool
        }
        __syncthreads();
    }
}

// ---------------------------------------------------------------------------
// 2) Edge features: x0[row, 0:64] = center, x0[row, 64:128] = knn - center
// ---------------------------------------------------------------------------
__global__ void edge_build(const float* __restrict__ F, const int* __restrict__ idx,
                           float* __restrict__ x0) {
    const size_t tid = (size_t)blockIdx.x * 256 + threadIdx.x;   // row*128 + c
    const int c   = (int)(tid & 127);
    const int row = (int)(tid >> 7);                // n*16384 + p*16 + k
    const int n   = row >> 14;
    const int p   = (row >> 4) & 1023;
    const int np  = n * P_PTS + p;
    float v;
    if (c < C0) {
        v = F[(size_t)np * C0 + c];
    } else {
        const int j  = idx[row];
        const int cc = c - C0;
        v = F[((size_t)n * P_PTS + j) * C0 + cc] - F[(size_t)np * C0 + cc];
    }
    x0[tid] = v;
}

// ---------------------------------------------------------------------------
// 3) GEMM via V_WMMA_F32_16X16X4_F32.
//    Block = 8 waves = 128-row M block x 16-col N tile.
//    A block (128 x CIN) is staged into LDS with coalesced async
//    global->LDS B128 copies (ASYNCcnt path); W tile also in LDS.
//    Inner loop is pure ds_load + wmma. Per-channel BN stats accumulated
//    via LDS then one global atomic per channel per block.
// ---------------------------------------------------------------------------
template <int CIN, int COUT>
__global__ __launch_bounds__(256) void gemm_wmma(const float* __restrict__ X,
                                                 const float* __restrict__ W,
                                                 float* __restrict__ Y,
                                                 float* __restrict__ stats) {
    constexpr int AST = CIN + 4;            // padded row stride: bank-conflict-free,
                                            // keeps 16B alignment (528B for CIN=128)
    __shared__ float aT[128 * AST];
    __shared__ float wT[CIN * 16];          // [k][c] for this 16-col tile
    __shared__ float sSum[16], sSq[16];

    const int wave = threadIdx.x >> 5;
    const int lane = threadIdx.x & 31;
    const int half = lane >> 4;             // 0: K{0,1} rows 0-7 | 1: K{2,3} rows 8-15
    const int l15  = lane & 15;
    const int n0   = blockIdx.y * 16;
    const size_t mBlk = (size_t)blockIdx.x * 128;

    // ---- coalesced async stage of the 128 x CIN A block into padded LDS ----
    constexpr int CHUNKS = 128 * CIN / 4;   // float4 chunks
    for (int i = threadIdx.x; i < CHUNKS; i += 256) {
        const int r  = i / (CIN / 4);
        const int c4 = (i % (CIN / 4)) * 4;
        const float* src = X + (mBlk + r) * CIN + c4;
        float* dst = &aT[r * AST + c4];
#if USE_ASYNC_LDS
        __builtin_amdgcn_global_load_async_to_lds_b128(
            (AS1 v4i*)src, (AS3 v4i*)dst, 0, 0);
#else
        *(v4f*)dst = *(const v4f*)src;
#endif
    }
    // ---- W column tile into LDS ----
    for (int i = threadIdx.x; i < CIN * 16; i += 256) {
        wT[i] = W[(size_t)(i >> 4) * COUT + n0 + (i & 15)];
    }
    if (threadIdx.x < 16) { sSum[threadIdx.x] = 0.f; sSq[threadIdx.x] = 0.f; }
#if USE_ASYNC_LDS
    __builtin_amdgcn_s_wait_asynccnt(0);
#endif
    __syncthreads();

    // ---- WMMA loop: everything from LDS ----
    const float* ar = &aT[(wave * 16 + l15) * AST + half * 2];
    v8f acc = {};
    #pragma unroll 4
    for (int k = 0; k < CIN; k += 4) {
        v2f a; a.x = ar[k];  a.y = ar[k + 1];
        v2f b; b.x = wT[(k + half * 2) * 16 + l15];
        b.y = wT[(k + half * 2 + 1) * 16 + l15];
        acc = __builtin_amdgcn_wmma_f32_16x16x4_f32(
            /*neg_a=*/false, a, /*neg_b=*/false, b,
            /*c_mod=*/(short)0, acc, /*reuse_a=*/false, /*reuse_b=*/false);
    }

    // ---- store + BN stats ----
    float s = 0.f, sq = 0.f;
    const size_t base = (mBlk + (size_t)wave * 16 + half * 8) * COUT + n0 + l15;
    #pragma unroll
    for (int i = 0; i < 8; ++i) {
        float v = acc[i];
        Y[base + (size_t)i * COUT] = v;
        s += v;
        sq += v * v;
    }
    atomicAdd(&sSum[l15], s);
    atomicAdd(&sSq[l15], sq);
    __syncthreads();
    if (threadIdx.x < 16) {
        atomicAdd(&stats[n0 + threadIdx.x],        sSum[threadIdx.x]);
        atomicAdd(&stats[COUT + n0 + threadIdx.x], sSq[threadIdx.x]);
    }
}

__device__ __forceinline__ float bn_lrelu_val(float x, float sum, float sq,
                                              float g, float b, float invCnt) {
    float mean = sum * invCnt;
    float var  = sq * invCnt - mean * mean;
    float sc   = g * __frsqrt_rn(var + BN_EPS);
    float v    = fmaf(x, sc, b - mean * sc);
    return v > 0.f ? v : ALPHA * v;
}

// 4) BN + LeakyReLU, in place
template <int COUT>
__global__ void bn_lrelu_inplace(float* __restrict__ buf, const float* __restrict__ stats,
                                 const float* __restrict__ g, const float* __restrict__ b,
                                 float invCnt) {
    const size_t tid = (size_t)blockIdx.x * 256 + threadIdx.x;
    const int c = (int)(tid & (COUT - 1));
    buf[tid] = bn_lrelu_val(buf[tid], stats[c], stats[COUT + c], g[c], b[c], invCnt);
}

// 5) Layer-3 BN + LeakyReLU fused with mean over K
__global__ void bn_lrelu_meanK(const float* __restrict__ y3, const float* __restrict__ stats,
                               const float* __restrict__ g, const float* __restrict__ b,
                               float* __restrict__ fts, float invCnt) {
    const int np = blockIdx.x;
    const int d  = threadIdx.x;           // 256 channels
    const float sum = stats[d], sq = stats[256 + d], gg = g[d], bb = b[d];
    float acc = 0.f;
    #pragma unroll
    for (int k = 0; k < K_NN; ++k) {
        acc += bn_lrelu_val(y3[((size_t)np * K_NN + k) * 256 + d], sum, sq, gg, bb, invCnt);
    }
    fts[(size_t)np * 256 + d] = acc * (1.0f / K_NN);
}

// 6) Shortcut BN + add + LeakyReLU
__global__ void final_combine(const float* __restrict__ ysc, const float* __restrict__ stats,
                              const float* __restrict__ g, const float* __restrict__ b,
                              const float* __restrict__ fts, float* __restrict__ out,
                              float invCnt) {
    const int np = blockIdx.x;
    const int d  = threadIdx.x;
    const size_t i = (size_t)np * 256 + d;
    float mean = stats[d] * invCnt;
    float var  = stats[256 + d] * invCnt - mean * mean;
    float sc   = g[d] * __frsqrt_rn(var + BN_EPS);
    float v    = fmaf(ysc[i], sc, b[d] - mean * sc) + fts[i];
    out[i] = v > 0.f ? v : ALPHA * v;
}

// ---------------------------------------------------------------------------
extern "C" void kernel_launch(void* const* d_in, const int* in_sizes, int n_in,
                              void* d_out, int out_size, void* d_ws, size_t ws_size,
                              hipStream_t stream) {
    const float* points   = (const float*)d_in[0];
    const float* features = (const float*)d_in[1];
    const float* W0 = (const float*)d_in[2];
    const float* g0 = (const float*)d_in[3];
    const float* b0 = (const float*)d_in[4];
    const float* W1 = (const float*)d_in[5];
    const float* g1 = (const float*)d_in[6];
    const float* b1 = (const float*)d_in[7];
    const float* W2 = (const float*)d_in[8];
    const float* g2 = (const float*)d_in[9];
    const float* b2 = (const float*)d_in[10];
    const float* Wsc = (const float*)d_in[11];
    const float* gsc = (const float*)d_in[12];
    const float* bsc = (const float*)d_in[13];
    float* out = (float*)d_out;

    const size_t M  = (size_t)N_B * P_PTS * K_NN;   // 524288 edge rows
    const size_t NP = (size_t)N_B * P_PTS;          // 32768

    // ---- workspace layout (bytes) ----
    char* ws = (char*)d_ws;
    int*   idx   = (int*)ws;                                       //  2 MB
    float* stats = (float*)(ws + (4u << 20));                      //  6 KB (4 blocks)
    float* st0   = stats;                                          // 2*128
    float* st1   = stats + 256;                                    // 2*128
    float* st2   = stats + 512;                                    // 2*256
    float* stsc  = stats + 1024;                                   // 2*256
    float* buf0  = (float*)(ws + (8u << 20));                      // M*128 f32 = 268 MB
    float* buf1  = buf0 + M * 128;                                 // 268 MB
    float* buf2  = buf1 + M * 128;                                 // M*256 f32 = 536 MB
    float* fts   = buf2 + M * 256;                                 // NP*256 = 33 MB
    float* ysc   = fts  + NP * 256;                                // 33 MB

    (void)hipMemsetAsync(stats, 0, 1536 * sizeof(float), stream);

    // 1) KNN indices
    knn_kernel<<<N_B * P_PTS, 128, 0, stream>>>(points, idx);

    // 2) Edge features x0: (M,128)
    edge_build<<<(unsigned)((M * 128) / 256), 256, 0, stream>>>(features, idx, buf0);

    // 3) Layer 0: (M,128)x(128,128)
    gemm_wmma<128, 128><<<dim3((unsigned)(M / 128), 8), 256, 0, stream>>>(buf0, W0, buf1, st0);
    bn_lrelu_inplace<128><<<(unsigned)((M * 128) / 256), 256, 0, stream>>>(
        buf1, st0, g0, b0, 1.0f / (float)M);

    // 4) Layer 1: (M,128)x(128,128)
    gemm_wmma<128, 128><<<dim3((unsigned)(M / 128), 8), 256, 0, stream>>>(buf1, W1, buf0, st1);
    bn_lrelu_inplace<128><<<(unsigned)((M * 128) / 256), 256, 0, stream>>>(
        buf0, st1, g1, b1, 1.0f / (float)M);

    // 5) Layer 2: (M,128)x(128,256), then BN+LReLU fused with mean over K
    gemm_wmma<128, 256><<<dim3((unsigned)(M / 128), 16), 256, 0, stream>>>(buf0, W2, buf2, st2);
    bn_lrelu_meanK<<<(unsigned)NP, 256, 0, stream>>>(buf2, st2, g2, b2, fts, 1.0f / (float)M);

    // 6) Shortcut: (NP,64)x(64,256) + BN + add + LReLU
    gemm_wmma<64, 256><<<dim3((unsigned)(NP / 128), 16), 256, 0, stream>>>(features, Wsc, ysc, stsc);
    final_combine<<<(unsigned)NP, 256, 0, stream>>>(ysc, stsc, gsc, bsc, fts, out,
                                                    1.0f / (float)NP);
}